// ConditionalSDFVAE_30107720745799
// MI455X (gfx1250) — compile-verified
//
#include <hip/hip_runtime.h>
#include <hip/hip_bf16.h>

// ---------------- problem constants ----------------
#define BATCH   65536
#define DIN     3
#define LDIM    64
#define HDIM    256
#define NEXP    8
#define KIN     67            // D + L
#define KPAD    68            // padded to multiple of 4 for WMMA K-steps
#define KG0     (KPAD / 4)    // 17 packed K-groups, layer 0
#define KGH     (HDIM / 4)    // 64 packed K-groups, hidden layers
#define MT      64            // rows per MoE block (4 M-tiles -> 4x B-fragment reuse)
#define LSTRIDE 258           // LDS row stride (floats): bank-conflict-free, 8B aligned

typedef __attribute__((ext_vector_type(2))) float v2f;
typedef __attribute__((ext_vector_type(8))) float v8f;

// ---------------- kernel 1: build x_in = [x | z | 0], count labels ----------------
__global__ void vae_prep_kernel(const float* __restrict__ x,
                                const long long* __restrict__ labels,
                                const float* __restrict__ eps,
                                const float* __restrict__ mu_t,
                                const float* __restrict__ lv_t,
                                float* __restrict__ xin,
                                int* __restrict__ counts)
{
    __shared__ int cnt[NEXP];
    if (threadIdx.x < NEXP) cnt[threadIdx.x] = 0;
    __syncthreads();

    int gid = blockIdx.x * blockDim.x + threadIdx.x;
    if (gid < BATCH * KPAD) {
        int b = gid / KPAD;
        int c = gid - b * KPAD;
        int lab = (int)labels[b];
        float v;
        if (c < DIN) {
            v = x[b * DIN + c];
        } else if (c < KIN) {
            int l = c - DIN;
            float lv = lv_t[lab * LDIM + l];
            v = mu_t[lab * LDIM + l] + eps[b * LDIM + l] * __expf(0.5f * lv);
        } else {
            v = 0.0f;
        }
        xin[b * KPAD + c] = v;
        if (c == 0) atomicAdd(&cnt[lab], 1);
    }
    __syncthreads();
    if (threadIdx.x < NEXP && cnt[threadIdx.x] > 0)
        atomicAdd(&counts[threadIdx.x], cnt[threadIdx.x]);
}

// ---------------- kernel 2: KL (deterministic, analytic over labels) + padded offsets ----------------
__global__ void vae_stats_kernel(const float* __restrict__ mu_t,
                                 const float* __restrict__ lv_t,
                                 const int* __restrict__ counts,
                                 int* __restrict__ offsets,
                                 int* __restrict__ cursors,
                                 float* __restrict__ kl_out)
{
    __shared__ float red[LDIM];
    int l = threadIdx.x;  // 64 threads
    float s = 0.0f;
    for (int e = 0; e < NEXP; ++e) {
        float lv = lv_t[e * LDIM + l];
        float mu = mu_t[e * LDIM + l];
        s += (float)counts[e] * (1.0f + lv - mu * mu - __expf(lv));
    }
    red[l] = s;
    __syncthreads();
    for (int off = LDIM / 2; off >= 1; off >>= 1) {
        if (l < off) red[l] += red[l + off];
        __syncthreads();
    }
    if (l == 0) {
        kl_out[0] = -0.5f * red[0] / (float)BATCH;
        int acc = 0;
        offsets[0] = 0;
        for (int e = 0; e < NEXP; ++e) {
            acc += (counts[e] + (MT - 1)) & ~(MT - 1);
            offsets[e + 1] = acc;
            cursors[e] = 0;
        }
    }
}

// ---------------- kernel 3: scatter rows into per-expert groups ----------------
__global__ void vae_scatter_kernel(const long long* __restrict__ labels,
                                   const int* __restrict__ offsets,
                                   int* __restrict__ cursors,
                                   int* __restrict__ gather)
{
    int b = blockIdx.x * blockDim.x + threadIdx.x;
    if (b < BATCH) {
        int e = (int)labels[b];
        int pos = offsets[e] + atomicAdd(&cursors[e], 1);
        gather[pos] = b;
    }
}

// ---------------- weight repack: W[K,N] -> Wp[K/4][N][4], K zero-padded ----------------
// Packed so a lane's B-fragment (rows k0,k0+1 at column n) is one contiguous 8B pair.
__global__ void vae_pack_w0_kernel(const float* __restrict__ W0, float* __restrict__ Wp0)
{
    int gid = blockIdx.x * blockDim.x + threadIdx.x;   // over NEXP*KG0*HDIM*4
    if (gid >= NEXP * KG0 * HDIM * 4) return;
    int j = gid & 3;
    int n = (gid >> 2) & (HDIM - 1);
    int g = (gid >> 10) % KG0;
    int e = gid / (KG0 * HDIM * 4);
    int k = 4 * g + j;
    Wp0[gid] = (k < KIN) ? W0[(e * KIN + k) * HDIM + n] : 0.0f;
}

__global__ void vae_pack_wh_kernel(const float* __restrict__ Wh, float* __restrict__ Wph)
{
    int gid = blockIdx.x * blockDim.x + threadIdx.x;   // over 2*NEXP*KGH*HDIM*4
    if (gid >= 2 * NEXP * KGH * HDIM * 4) return;
    int j = gid & 3;
    int n = (gid >> 2) & (HDIM - 1);
    int g = (gid >> 10) & (KGH - 1);
    int le = gid / (KGH * HDIM * 4);                   // l*NEXP + e
    int k = 4 * g + j;
    Wph[gid] = Wh[(le * HDIM + k) * HDIM + n];
}

// ---------------- fused relu-MLP layer: h[MT,Kp] (LDS) x packed Wp -> h[MT,H] ----------------
template <int Kp>
__device__ __forceinline__ void mlp_layer_wmma(float* hbuf,
                                               const float* __restrict__ Wp,
                                               const float* __restrict__ bias)
{
    const int tid   = threadIdx.x;
    const int wv    = tid >> 5;          // 8 waves
    const int lane  = tid & 31;
    const int mrow  = lane & 15;         // A-matrix M row
    const int khalf = lane >> 4;         // K sub-pair select {0,1}
    const int koff  = khalf << 1;        // A K sub-offset {0,2}
    const int ncol  = lane & 15;         // B/C/D N column

    const v2f* __restrict__ Wp2 = (const v2f*)Wp;   // 8B pairs

    v8f acc[4][2] = {};  // [m-tile][col-tile]

    for (int k = 0; k < Kp; k += 4) {
        const int k0 = k + koff;
        const int k1 = k0 + 1;
        v2f a[4];
#pragma unroll
        for (int mt = 0; mt < 4; ++mt) {
            a[mt].x = hbuf[(mt * 16 + mrow) * LSTRIDE + k0];
            a[mt].y = hbuf[(mt * 16 + mrow) * LSTRIDE + k1];
        }
#pragma unroll
        for (int ct = 0; ct < 2; ++ct) {
            const int colbase = (wv + ct * 8) * 16;
            // one coalesced b64 load per fragment
            const v2f bf = Wp2[((k >> 2) * HDIM + colbase + ncol) * 2 + khalf];
#pragma unroll
            for (int mt = 0; mt < 4; ++mt) {
                acc[mt][ct] = __builtin_amdgcn_wmma_f32_16x16x4_f32(
                    false, a[mt], false, bf, (short)0, acc[mt][ct], false, false);
            }
        }
    }
    __syncthreads();   // all reads of hbuf complete before overwrite
#pragma unroll
    for (int ct = 0; ct < 2; ++ct) {
        const int colbase = (wv + ct * 8) * 16;
        const float bv = bias[colbase + ncol];
#pragma unroll
        for (int mt = 0; mt < 4; ++mt) {
#pragma unroll
            for (int r = 0; r < 8; ++r) {
                float v = acc[mt][ct][r] + bv;
                v = v > 0.0f ? v : 0.0f;   // relu
                const int M = mt * 16 + r + 8 * khalf;
                hbuf[M * LSTRIDE + colbase + ncol] = v;
            }
        }
    }
    __syncthreads();
}

// ---------------- kernel 4: grouped-expert MLP via WMMA ----------------
__global__ void __launch_bounds__(256)
vae_moe_kernel(const float* __restrict__ xin,
               const int* __restrict__ gather,
               const int* __restrict__ offsets,
               const float* __restrict__ Wp0, const float* __restrict__ b0,
               const float* __restrict__ Wph, const float* __restrict__ bh,
               const float* __restrict__ Wo,  const float* __restrict__ bo,
               float* __restrict__ out)
{
    extern __shared__ float hbuf[];     // MT * LSTRIDE floats (dynamic: > 64KB static limit)
    __shared__ int gsh[MT];
    __shared__ int esh;

    const int base = blockIdx.x * MT;
    if (threadIdx.x == 0) {
        int e = -1;
        if (base < offsets[NEXP]) {
            e = 0;
            while (base >= offsets[e + 1]) ++e;
        }
        esh = e;
    }
    __syncthreads();
    const int e = esh;
    if (e < 0) return;   // uniform exit: padding tiles past total

    if (threadIdx.x < MT) gsh[threadIdx.x] = gather[base + threadIdx.x];
    __syncthreads();

    // stage gathered x_in rows into LDS (padding rows -> zeros)
    for (int i = threadIdx.x; i < MT * KPAD; i += 256) {
        const int r = i / KPAD;
        const int c = i - r * KPAD;
        const int g = gsh[r];
        hbuf[r * LSTRIDE + c] = (g >= 0) ? xin[g * KPAD + c] : 0.0f;
    }
    __syncthreads();

    // 67 -> 256 -> 256 -> 256 (relu each), packed weights
    mlp_layer_wmma<KPAD>(hbuf, Wp0 + e * KG0 * HDIM * 4,              b0 + e * HDIM);
    mlp_layer_wmma<HDIM>(hbuf, Wph + (0 * NEXP + e) * KGH * HDIM * 4, bh + (0 * NEXP + e) * HDIM);
    mlp_layer_wmma<HDIM>(hbuf, Wph + (1 * NEXP + e) * KGH * HDIM * 4, bh + (1 * NEXP + e) * HDIM);

    // 256 -> 1 (no relu): 4 lanes per row (64 rows), shuffle reduce
    const int t  = threadIdx.x;
    const int r  = t >> 2;          // 0..63
    const int c0 = (t & 3) * 64;
    float p = 0.0f;
#pragma unroll 8
    for (int j = 0; j < 64; ++j)
        p += hbuf[r * LSTRIDE + c0 + j] * Wo[e * HDIM + c0 + j];
    p += __shfl_xor(p, 2, 32);
    p += __shfl_xor(p, 1, 32);
    if ((t & 3) == 0) {
        const int g = gsh[r];
        if (g >= 0) out[g] = p + bo[e];
    }
}

// ---------------- host launch ----------------
extern "C" void kernel_launch(void* const* d_in, const int* in_sizes, int n_in,
                              void* d_out, int out_size, void* d_ws, size_t ws_size,
                              hipStream_t stream)
{
    const float*     x      = (const float*)d_in[0];
    const long long* labels = (const long long*)d_in[1];
    const float*     eps    = (const float*)d_in[2];
    const float*     mu_t   = (const float*)d_in[3];
    const float*     lv_t   = (const float*)d_in[4];
    const float*     W0     = (const float*)d_in[5];
    const float*     b0     = (const float*)d_in[6];
    const float*     Wh     = (const float*)d_in[7];
    const float*     bh     = (const float*)d_in[8];
    const float*     Wo     = (const float*)d_in[9];
    const float*     bo     = (const float*)d_in[10];
    float* out = (float*)d_out;          // [BATCH] outputs + [1] KL at the end

    // workspace layout
    char* ws = (char*)d_ws;
    size_t off = 0;
    float* xin = (float*)(ws + off);  off += (size_t)BATCH * KPAD * sizeof(float);
    int* counts  = (int*)(ws + off);  off += 64 * sizeof(int);   // counts[8] | offsets[9] | cursors[8] padded
    int* offsets = counts + 16;
    int* cursors = counts + 32;
    int* gather  = (int*)(ws + off);  off += (size_t)(BATCH + NEXP * MT) * sizeof(int);
    float* Wp0   = (float*)(ws + off); off += (size_t)NEXP * KG0 * HDIM * 4 * sizeof(float);
    float* Wph   = (float*)(ws + off); off += (size_t)2 * NEXP * KGH * HDIM * 4 * sizeof(float);

    const int gather_n = BATCH + NEXP * MT;

    hipMemsetAsync(counts, 0, 8 * sizeof(int), stream);
    hipMemsetAsync(gather, 0xFF, (size_t)gather_n * sizeof(int), stream);  // -1 sentinels

    {
        const int total = BATCH * KPAD;
        vae_prep_kernel<<<(total + 255) / 256, 256, 0, stream>>>(
            x, labels, eps, mu_t, lv_t, xin, counts);
    }
    {
        const int n0 = NEXP * KG0 * HDIM * 4;
        vae_pack_w0_kernel<<<(n0 + 255) / 256, 256, 0, stream>>>(W0, Wp0);
        const int nh = 2 * NEXP * KGH * HDIM * 4;
        vae_pack_wh_kernel<<<(nh + 255) / 256, 256, 0, stream>>>(Wh, Wph);
    }
    vae_stats_kernel<<<1, LDIM, 0, stream>>>(mu_t, lv_t, counts, offsets, cursors, out + BATCH);
    vae_scatter_kernel<<<BATCH / 256, 256, 0, stream>>>(labels, offsets, cursors, gather);

    const size_t lds_bytes = (size_t)MT * LSTRIDE * sizeof(float);   // 66,048 B dynamic LDS
    hipFuncSetAttribute((const void*)vae_moe_kernel,
                        hipFuncAttributeMaxDynamicSharedMemorySize, (int)lds_bytes);

    const int ntiles = (BATCH + NEXP * MT) / MT;     // 1032, extras early-exit
    vae_moe_kernel<<<ntiles, 256, lds_bytes, stream>>>(
        xin, gather, offsets, Wp0, b0, Wph, bh, Wo, bo, out);
}